// WindowEmbedding_7086696038878
// MI455X (gfx1250) — compile-verified
//
#include <hip/hip_runtime.h>
#include <hip/hip_bf16.h>

// WindowEmbedding: out[b, t, i*D + d] = (t+i < T) ? x[b, t+i, d] : 0
// x: [B=16, T=2048, D=512] f32 -> out: [B, T, D*W] f32, W=5.
//
// Pure data movement (0 FLOPs). Min HBM traffic: 64 MiB read + 320 MiB write
// ~= 403 MB -> ~17 us at 23.3 TB/s peak. x (64 MiB) fits in the 192 MB L2, so
// the 5x logical re-read is L2-served; the 320 MiB output stream is marked
// non-temporal so it does not evict x.
//
// Structure: fast path (t+W <= T, 99.8% of rows) issues 5 independent
// global_load_b128 back-to-back (max MLP per wave), then 5 NT stores; the
// rare sequence-tail path applies the per-slot zero guard.

typedef float __attribute__((ext_vector_type(4))) f4;

namespace {
constexpr int W      = 5;
constexpr int T      = 2048;           // power of two -> mask instead of mod
constexpr int D      = 512;
constexpr int D4     = D / 4;          // 128 float4 per row-slot (power of two)
constexpr int D4_SH  = 7;              // log2(D4)
constexpr int DW4    = W * D4;         // 640 float4 per output row
}

__global__ __launch_bounds__(256)
void window_embed_b128_nt(const f4* __restrict__ x,
                          f4* __restrict__ out,
                          int n_rows /* B*T */) {
  const int idx = blockIdx.x * blockDim.x + threadIdx.x;   // (row, d4) pair
  const int d4  = idx & (D4 - 1);
  const int row = idx >> D4_SH;
  if (row >= n_rows) return;

  const int t = row & (T - 1);                 // position within the sequence
  const long out_base = (long)row * DW4 + d4;  // float4 units
  const long in_base  = (long)row * D4 + d4;   // float4 units

  if (t + W <= T) {
    // Fast path: all 5 source rows valid. Issue all loads first so the wave
    // keeps 5 b128 loads in flight, then drain into 5 NT b128 stores.
    f4 v0 = x[in_base + 0 * D4];
    f4 v1 = x[in_base + 1 * D4];
    f4 v2 = x[in_base + 2 * D4];
    f4 v3 = x[in_base + 3 * D4];
    f4 v4 = x[in_base + 4 * D4];
    __builtin_nontemporal_store(v0, &out[out_base + 0 * D4]);
    __builtin_nontemporal_store(v1, &out[out_base + 1 * D4]);
    __builtin_nontemporal_store(v2, &out[out_base + 2 * D4]);
    __builtin_nontemporal_store(v3, &out[out_base + 3 * D4]);
    __builtin_nontemporal_store(v4, &out[out_base + 4 * D4]);
  } else {
    // Sequence tail (last W-1 rows of each batch): per-slot zero padding.
    const f4 zero = {0.0f, 0.0f, 0.0f, 0.0f};
#pragma unroll
    for (int i = 0; i < W; ++i) {
      f4 v = zero;
      if (t + i < T) v = x[in_base + (long)i * D4];
      __builtin_nontemporal_store(v, &out[out_base + (long)i * D4]);
    }
  }
}

extern "C" void kernel_launch(void* const* d_in, const int* in_sizes, int n_in,
                              void* d_out, int out_size, void* d_ws, size_t ws_size,
                              hipStream_t stream) {
  const f4* x  = (const f4*)d_in[0];
  f4* out      = (f4*)d_out;

  const int n_rows  = in_sizes[0] / D;         // B*T = 32768
  const int n_items = n_rows << D4_SH;         // one thread per (row, d4)
  const int block   = 256;                     // 8 wave32 per block
  const int grid    = (n_items + block - 1) / block;

  window_embed_b128_nt<<<grid, block, 0, stream>>>(x, out, n_rows);
}